// GAT_42984032698810
// MI455X (gfx1250) — compile-verified
//
#include <hip/hip_runtime.h>
#include <math.h>

#define NNODES 50000
#define NEDGES 800000
#define NGRAPH 64
#define NH 16
#define NC 4
#define HC 64   // NH*NC

typedef __attribute__((ext_vector_type(2))) float v2f;
typedef __attribute__((ext_vector_type(8))) float v8f;

// ---------------------------------------------------------------- utilities

__global__ void k_fill(float* p, float v, int n) {
    int i = blockIdx.x * blockDim.x + threadIdx.x;
    if (i < n) p[i] = v;
}

__device__ inline void atomicMaxF(float* addr, float val) {
    int* ai = (int*)addr;
    int old = __float_as_int(*addr);
    while (__int_as_float(old) < val) {
        int assumed = old;
        old = atomicCAS(ai, assumed, __float_as_int(val));
        if (old == assumed) break;
    }
}

// ---------------------------------------------------------------- GAT front

// xl = x*Wl + bl ; xr = x*Wr + br   (per node, 64 channels)
__global__ void k_project(const float* __restrict__ x,
                          const float* __restrict__ Wl, const float* __restrict__ bl,
                          const float* __restrict__ Wr, const float* __restrict__ br,
                          float* __restrict__ xl, float* __restrict__ xr) {
    int i = blockIdx.x * blockDim.x + threadIdx.x;
    if (i >= NNODES * HC) return;
    int node = i >> 6, j = i & 63;
    float xv = x[node];
    xl[i] = xv * Wl[j] + bl[j];
    xr[i] = xv * Wr[j] + br[j];
}

// in-degree and self-loop attribute accumulation
__global__ void k_deg(const int* __restrict__ ei, const float* __restrict__ ea,
                      float* __restrict__ deg, float* __restrict__ lsum) {
    int e = blockIdx.x * blockDim.x + threadIdx.x;
    if (e >= NEDGES) return;
    int dst = ei[NEDGES + e];
    atomicAdd(&deg[dst], 1.0f);
    atomicAdd(&lsum[dst], ea[e]);
}

__global__ void k_loopattr(const float* __restrict__ deg, const float* __restrict__ lsum,
                           float* __restrict__ lattr) {
    int i = blockIdx.x * blockDim.x + threadIdx.x;
    if (i >= NNODES) return;
    lattr[i] = lsum[i] / fmaxf(deg[i], 1.0f);
}

// per (augmented) edge: 16 attention scores + running segment max
__global__ void k_score(const int* __restrict__ ei, const float* __restrict__ ea,
                        const float* __restrict__ lattr,
                        const float* __restrict__ xl, const float* __restrict__ xr,
                        const float* __restrict__ We, const float* __restrict__ att,
                        float* __restrict__ score, float* __restrict__ smax) {
    int e = blockIdx.x * blockDim.x + threadIdx.x;
    if (e >= NEDGES + NNODES) return;
    int src, dst; float a;
    if (e < NEDGES) { src = ei[e]; dst = ei[NEDGES + e]; a = ea[e]; }
    else            { src = e - NEDGES; dst = src; a = lattr[src]; }
    const float* pl = xl + (size_t)src * HC;
    const float* pr = xr + (size_t)dst * HC;
    for (int h = 0; h < NH; ++h) {
        float s = 0.f;
        for (int c = 0; c < NC; ++c) {
            int j = h * NC + c;
            float m = pl[j] + pr[j] + a * We[j];
            m = (m > 0.f) ? m : 0.2f * m;          // leaky_relu 0.2
            s += m * att[j];
        }
        score[(size_t)e * NH + h] = s;
        atomicMaxF(&smax[dst * NH + h], s);
    }
}

// a = exp(score - smax[dst]); denom += a   (stored in place of score)
__global__ void k_exp(const int* __restrict__ ei, float* __restrict__ score,
                      const float* __restrict__ smax, float* __restrict__ denom) {
    int e = blockIdx.x * blockDim.x + threadIdx.x;
    if (e >= NEDGES + NNODES) return;
    int dst = (e < NEDGES) ? ei[NEDGES + e] : (e - NEDGES);
    for (int h = 0; h < NH; ++h) {
        float v = __expf(score[(size_t)e * NH + h] - smax[dst * NH + h]);
        score[(size_t)e * NH + h] = v;
        atomicAdd(&denom[dst * NH + h], v);
    }
}

// h[dst] += alpha * xl[src]
__global__ void k_agg(const int* __restrict__ ei, const float* __restrict__ score,
                      const float* __restrict__ denom, const float* __restrict__ xl,
                      float* __restrict__ hbuf) {
    int e = blockIdx.x * blockDim.x + threadIdx.x;
    if (e >= NEDGES + NNODES) return;
    int src, dst;
    if (e < NEDGES) { src = ei[e]; dst = ei[NEDGES + e]; }
    else            { src = e - NEDGES; dst = src; }
    const float* pl = xl + (size_t)src * HC;
    float* pd = hbuf + (size_t)dst * HC;
    for (int h = 0; h < NH; ++h) {
        float alpha = score[(size_t)e * NH + h] / denom[dst * NH + h];
        for (int c = 0; c < NC; ++c) {
            int j = h * NC + c;
            atomicAdd(&pd[j], alpha * pl[j]);
        }
    }
}

// ---------------------------------------------------------------- GroupNorm

__global__ void k_cnt(const int* __restrict__ batch, float* __restrict__ cnt) {
    int i = blockIdx.x * blockDim.x + threadIdx.x;
    if (i >= NNODES) return;
    atomicAdd(&cnt[batch[i]], 1.0f);
}

// h = relu(h + gat_bias); gsum[batch] += h
__global__ void k_relu_bias_sum(float* __restrict__ hbuf, const float* __restrict__ gat_bias,
                                const int* __restrict__ batch, float* __restrict__ gsum) {
    int i = blockIdx.x * blockDim.x + threadIdx.x;
    if (i >= NNODES * HC) return;
    int node = i >> 6, j = i & 63;
    float v = fmaxf(hbuf[i] + gat_bias[j], 0.0f);
    hbuf[i] = v;
    atomicAdd(&gsum[batch[node] * HC + j], v);
}

__global__ void k_gmean(const float* __restrict__ gsum, const float* __restrict__ cnt,
                        float* __restrict__ gmean) {
    int i = blockIdx.x * blockDim.x + threadIdx.x;
    if (i >= NGRAPH * HC) return;
    gmean[i] = gsum[i] / cnt[i >> 6];
}

__global__ void k_var(const float* __restrict__ hbuf, const int* __restrict__ batch,
                      const float* __restrict__ gmean, const float* __restrict__ gn_ms,
                      float* __restrict__ gvar) {
    int i = blockIdx.x * blockDim.x + threadIdx.x;
    if (i >= NNODES * HC) return;
    int node = i >> 6, j = i & 63;
    float o = hbuf[i] - gmean[batch[node] * HC + j] * gn_ms[j];
    atomicAdd(&gvar[batch[node] * HC + j], o * o);
}

__global__ void k_gnorm(float* __restrict__ hbuf, const int* __restrict__ batch,
                        const float* __restrict__ gmean, const float* __restrict__ gn_ms,
                        const float* __restrict__ gvar, const float* __restrict__ cnt,
                        const float* __restrict__ gn_w, const float* __restrict__ gn_b) {
    int i = blockIdx.x * blockDim.x + threadIdx.x;
    if (i >= NNODES * HC) return;
    int node = i >> 6, j = i & 63;
    int b = batch[node];
    float o = hbuf[i] - gmean[b * HC + j] * gn_ms[j];
    float var = gvar[b * HC + j] / cnt[b];
    hbuf[i] = gn_w[j] * o * rsqrtf(var + 1e-5f) + gn_b[j];
}

// ---------------------------------------------------------------- SAGE gather

__global__ void k_sage_gather(const int* __restrict__ ei, const float* __restrict__ hbuf,
                              float* __restrict__ neigh) {
    int i = blockIdx.x * blockDim.x + threadIdx.x;   // E*64 = 51.2M < 2^31
    if (i >= NEDGES * HC) return;
    int e = i >> 6, j = i & 63;
    atomicAdd(&neigh[(size_t)ei[NEDGES + e] * HC + j], hbuf[(size_t)ei[e] * HC + j]);
}

__global__ void k_sage_div(float* __restrict__ neigh, const float* __restrict__ deg) {
    int i = blockIdx.x * blockDim.x + threadIdx.x;
    if (i >= NNODES * HC) return;
    neigh[i] /= fmaxf(deg[i >> 6], 1.0f);
}

// ---------------------------------------------------------------- fp32 WMMA GEMM
// C[M x Nw] = relu?( A[M x K] @ B[K x Nw] (+ A2 @ B2) + bias )
// Block = 8 waves, each wave owns one 16x16 output tile of one M-chunk;
// whole block shares one 16-column B panel staged in LDS in 128-row chunks.
// A-row index is clamped (not masked) so the WMMA inner loop is branch-free
// and EXEC stays all-1s; out-of-range rows are masked only at the store.
template<bool DUAL>
__global__ __launch_bounds__(256) void
k_wmma_gemm(const float* __restrict__ A, const float* __restrict__ B,
            const float* __restrict__ A2, const float* __restrict__ B2,
            const float* __restrict__ bias, float* __restrict__ C,
            int M, int K, int Nw, int relu) {
    const int KC = 128;                 // K rows of B staged per chunk
    __shared__ float Bs[KC * 16];
    __shared__ float Bs2[KC * 16];      // second panel for the fused dual GEMM

    int wid  = threadIdx.x >> 5;
    int lane = threadIdx.x & 31;
    int tm = blockIdx.x * 8 + wid;      // M-tile of this wave
    int tn = blockIdx.y;                // N-tile (shared by the block)
    int tilesM = (M + 15) >> 4;
    bool active = (tm < tilesM);        // wave-uniform
    int row0 = tm << 4, col0 = tn << 4;
    int half = lane >> 4, l16 = lane & 15;
    int m = row0 + l16; if (m > M - 1) m = M - 1;   // clamp: garbage rows masked at store
    int n = col0 + l16;

    v8f acc = {0.f, 0.f, 0.f, 0.f, 0.f, 0.f, 0.f, 0.f};

    for (int kc = 0; kc < K; kc += KC) {
        int chunk = K - kc; if (chunk > KC) chunk = KC;
        __syncthreads();
        for (int t = threadIdx.x; t < chunk * 16; t += 256) {
            int kk = t >> 4, nn = t & 15;
            Bs[t] = B[(size_t)(kc + kk) * Nw + col0 + nn];
            if (DUAL) Bs2[t] = B2[(size_t)(kc + kk) * Nw + col0 + nn];
        }
        __syncthreads();
        if (active) {
            const float* Ap  = A + (size_t)m * K + kc;
            const float* Ap2 = DUAL ? (A2 + (size_t)m * K + kc) : nullptr;
            for (int k0 = 0; k0 < chunk; k0 += 4) {
                int ka = k0 + half * 2;  // lanes 0-15: K={k0,k0+1}; 16-31: K={k0+2,k0+3}
                v2f av = *(const v2f*)(Ap + ka);           // one 8B global load
                v2f bv;
                bv.x = Bs[ka * 16 + l16];                  // LDS hits
                bv.y = Bs[(ka + 1) * 16 + l16];
                acc = __builtin_amdgcn_wmma_f32_16x16x4_f32(false, av, false, bv,
                                                            (short)0, acc, false, false);
                if (DUAL) {
                    v2f av2 = *(const v2f*)(Ap2 + ka);
                    v2f bv2;
                    bv2.x = Bs2[ka * 16 + l16];
                    bv2.y = Bs2[(ka + 1) * 16 + l16];
                    acc = __builtin_amdgcn_wmma_f32_16x16x4_f32(false, av2, false, bv2,
                                                                (short)0, acc, false, false);
                }
            }
        }
    }

    if (active) {
        float bb = bias ? bias[n] : 0.f;
        // C/D layout: VGPR v -> row = row0 + v + (lane>=16 ? 8 : 0), col = col0 + (lane&15)
        for (int v = 0; v < 8; ++v) {
            int r = row0 + v + half * 8;
            if (r < M) {
                float val = acc[v] + bb;
                if (relu) val = fmaxf(val, 0.f);
                C[(size_t)r * Nw + n] = val;
            }
        }
    }
}

// ---------------------------------------------------------------- pooling + fc3

__global__ void k_pool(const float* __restrict__ h2, const int* __restrict__ batch,
                       float* __restrict__ gmaxb, float* __restrict__ gsumb) {
    int i = blockIdx.x * blockDim.x + threadIdx.x;
    if (i >= NNODES * HC) return;
    int node = i >> 6, j = i & 63;
    int b = batch[node];
    float v = h2[i];
    atomicMaxF(&gmaxb[b * HC + j], v);
    atomicAdd(&gsumb[b * HC + j], v);
}

__global__ void k_pool_final(const float* __restrict__ gmaxb, const float* __restrict__ gsumb,
                             const float* __restrict__ cnt, float* __restrict__ g) {
    int i = blockIdx.x * blockDim.x + threadIdx.x;
    if (i >= NGRAPH * HC) return;
    int b = i >> 6, j = i & 63;
    g[b * 128 + j]      = gmaxb[i];
    g[b * 128 + 64 + j] = gsumb[i] / cnt[b];
}

__global__ void k_fc3(const float* __restrict__ act2, const float* __restrict__ W,
                      const float* __restrict__ b, float* __restrict__ out) {
    int i = blockIdx.x * blockDim.x + threadIdx.x;
    if (i >= NGRAPH * 3) return;
    int g = i / 3, o = i - g * 3;
    float s = b[o];
    for (int k = 0; k < 1024; ++k) s += act2[g * 1024 + k] * W[k * 3 + o];
    out[i] = s;
}

// ---------------------------------------------------------------- launch

static inline void fillf(float* p, float v, int n, hipStream_t s) {
    hipLaunchKernelGGL(k_fill, dim3((n + 255) / 256), dim3(256), 0, s, p, v, n);
}

extern "C" void kernel_launch(void* const* d_in, const int* in_sizes, int n_in,
                              void* d_out, int out_size, void* d_ws, size_t ws_size,
                              hipStream_t stream) {
    const float* x        = (const float*)d_in[0];
    const int*   ei       = (const int*)d_in[1];
    const float* ea       = (const float*)d_in[2];
    const int*   batch    = (const int*)d_in[3];
    const float* Wl       = (const float*)d_in[4];
    const float* bl       = (const float*)d_in[5];
    const float* Wr       = (const float*)d_in[6];
    const float* br       = (const float*)d_in[7];
    const float* We       = (const float*)d_in[8];
    const float* att      = (const float*)d_in[9];
    const float* gat_bias = (const float*)d_in[10];
    const float* gn_w     = (const float*)d_in[11];
    const float* gn_b     = (const float*)d_in[12];
    const float* gn_ms    = (const float*)d_in[13];
    const float* sage_Wl  = (const float*)d_in[14];
    const float* sage_bl  = (const float*)d_in[15];
    const float* sage_Wr  = (const float*)d_in[16];
    const float* fc1_W    = (const float*)d_in[17];
    const float* fc1_b    = (const float*)d_in[18];
    const float* fc2_W    = (const float*)d_in[19];
    const float* fc2_b    = (const float*)d_in[20];
    const float* fc3_W    = (const float*)d_in[21];
    const float* fc3_b    = (const float*)d_in[22];
    float* out = (float*)d_out;

    // -------- workspace layout (floats), ~127 MB total
    float* ws = (float*)d_ws;
    size_t off = 0;
    float* xl    = ws + off; off += (size_t)NNODES * HC;
    float* xr    = ws + off; off += (size_t)NNODES * HC;
    float* hbuf  = ws + off; off += (size_t)NNODES * HC;
    float* neigh = ws + off; off += (size_t)NNODES * HC;
    float* h2    = ws + off; off += (size_t)NNODES * HC;
    float* score = ws + off; off += (size_t)(NEDGES + NNODES) * NH;
    float* smax  = ws + off; off += (size_t)NNODES * NH;
    float* denom = ws + off; off += (size_t)NNODES * NH;
    float* deg   = ws + off; off += NNODES;
    float* lsum  = ws + off; off += NNODES;
    float* lattr = ws + off; off += NNODES;
    float* cnt   = ws + off; off += NGRAPH;
    float* gsum  = ws + off; off += NGRAPH * HC;
    float* gvar  = ws + off; off += NGRAPH * HC;
    float* gmean = ws + off; off += NGRAPH * HC;
    float* gmaxb = ws + off; off += NGRAPH * HC;
    float* gsumb = ws + off; off += NGRAPH * HC;
    float* gpool = ws + off; off += NGRAPH * 128;
    float* act1  = ws + off; off += NGRAPH * 2048;
    float* act2  = ws + off; off += NGRAPH * 1024;
    (void)ws_size; (void)in_sizes; (void)n_in; (void)out_size;

    const int NB = 256;
    const float NEGBIG = -3.402823466e38f;

    // -------- init accumulators
    fillf(deg,   0.f, NNODES, stream);
    fillf(lsum,  0.f, NNODES, stream);
    fillf(smax,  NEGBIG, NNODES * NH, stream);
    fillf(denom, 0.f, NNODES * NH, stream);
    fillf(hbuf,  0.f, NNODES * HC, stream);
    fillf(neigh, 0.f, NNODES * HC, stream);
    fillf(cnt,   0.f, NGRAPH, stream);
    fillf(gsum,  0.f, NGRAPH * HC, stream);
    fillf(gvar,  0.f, NGRAPH * HC, stream);
    fillf(gmaxb, NEGBIG, NGRAPH * HC, stream);
    fillf(gsumb, 0.f, NGRAPH * HC, stream);

    // -------- GAT
    hipLaunchKernelGGL(k_project, dim3((NNODES * HC + NB - 1) / NB), dim3(NB), 0, stream,
                       x, Wl, bl, Wr, br, xl, xr);
    hipLaunchKernelGGL(k_deg, dim3((NEDGES + NB - 1) / NB), dim3(NB), 0, stream,
                       ei, ea, deg, lsum);
    hipLaunchKernelGGL(k_loopattr, dim3((NNODES + NB - 1) / NB), dim3(NB), 0, stream,
                       deg, lsum, lattr);
    int EN = NEDGES + NNODES;
    hipLaunchKernelGGL(k_score, dim3((EN + NB - 1) / NB), dim3(NB), 0, stream,
                       ei, ea, lattr, xl, xr, We, att, score, smax);
    hipLaunchKernelGGL(k_exp, dim3((EN + NB - 1) / NB), dim3(NB), 0, stream,
                       ei, score, smax, denom);
    hipLaunchKernelGGL(k_agg, dim3((EN + NB - 1) / NB), dim3(NB), 0, stream,
                       ei, score, denom, xl, hbuf);

    // -------- GroupNorm
    hipLaunchKernelGGL(k_cnt, dim3((NNODES + NB - 1) / NB), dim3(NB), 0, stream, batch, cnt);
    hipLaunchKernelGGL(k_relu_bias_sum, dim3((NNODES * HC + NB - 1) / NB), dim3(NB), 0, stream,
                       hbuf, gat_bias, batch, gsum);
    hipLaunchKernelGGL(k_gmean, dim3((NGRAPH * HC + NB - 1) / NB), dim3(NB), 0, stream,
                       gsum, cnt, gmean);
    hipLaunchKernelGGL(k_var, dim3((NNODES * HC + NB - 1) / NB), dim3(NB), 0, stream,
                       hbuf, batch, gmean, gn_ms, gvar);
    hipLaunchKernelGGL(k_gnorm, dim3((NNODES * HC + NB - 1) / NB), dim3(NB), 0, stream,
                       hbuf, batch, gmean, gn_ms, gvar, cnt, gn_w, gn_b);

    // -------- SAGE neighbor mean
    hipLaunchKernelGGL(k_sage_gather, dim3((NEDGES * HC + NB - 1) / NB), dim3(NB), 0, stream,
                       ei, hbuf, neigh);
    hipLaunchKernelGGL(k_sage_div, dim3((NNODES * HC + NB - 1) / NB), dim3(NB), 0, stream,
                       neigh, deg);

    // -------- SAGE fused dual GEMM: h2 = relu(neigh@Wl + hbuf@Wr + bl)   [fp32 WMMA]
    {
        int tilesM = (NNODES + 15) / 16;                       // 3125
        dim3 grid((tilesM + 7) / 8, HC / 16);                  // 391 x 4
        k_wmma_gemm<true><<<grid, dim3(NB), 0, stream>>>(
            neigh, sage_Wl, hbuf, sage_Wr, sage_bl, h2, NNODES, HC, HC, 1);
    }

    // -------- pooling -> gpool (64 x 128)
    hipLaunchKernelGGL(k_pool, dim3((NNODES * HC + NB - 1) / NB), dim3(NB), 0, stream,
                       h2, batch, gmaxb, gsumb);
    hipLaunchKernelGGL(k_pool_final, dim3((NGRAPH * HC + NB - 1) / NB), dim3(NB), 0, stream,
                       gmaxb, gsumb, cnt, gpool);

    // -------- MLP head: fc1 (64x128 @ 128x2048), fc2 (64x2048 @ 2048x1024) [fp32 WMMA]
    {
        dim3 grid(1, 2048 / 16);                               // 4 M-tiles in one block
        k_wmma_gemm<false><<<grid, dim3(NB), 0, stream>>>(
            gpool, fc1_W, nullptr, nullptr, fc1_b, act1, NGRAPH, 128, 2048, 1);
    }
    {
        dim3 grid(1, 1024 / 16);
        k_wmma_gemm<false><<<grid, dim3(NB), 0, stream>>>(
            act1, fc2_W, nullptr, nullptr, fc2_b, act2, NGRAPH, 2048, 1024, 1);
    }

    // -------- fc3 (N=3, scalar kernel) -> d_out (64 x 3)
    hipLaunchKernelGGL(k_fc3, dim3((NGRAPH * 3 + NB - 1) / NB), dim3(NB), 0, stream,
                       act2, fc3_W, fc3_b, out);
}